// OpTokGen_11510512353892
// MI455X (gfx1250) — compile-verified
//
#include <hip/hip_runtime.h>
#include <stdint.h>

// ---------------------------------------------------------------------------
// attn/softattn/logPs for n-best rescoring:
//   logPs[s]   = sum_j log_theta[ids[s][j]]            (padding idx -> 0.0)
//   softattn   = softmax over groups of m=8
//   attn       = temp-softmax (tau=0.1) over groups of m=8
// Memory-bound: 64 MiB of ids dominate -> ~2.9us floor @ 23.3 TB/s.
// CDNA5 path: TDM-stage the 125 KiB table into 320KB LDS, ds_load_b32 gathers,
// wave32 8-lane shfl_xor softmax.
// ---------------------------------------------------------------------------

typedef unsigned int u32x4 __attribute__((ext_vector_type(4)));
typedef int          i32x8 __attribute__((ext_vector_type(8)));
typedef int          i32x4 __attribute__((ext_vector_type(4)));

#define TBL      32002      // V + 2 (log_theta with [-inf, 0.0] padding tail)
#define SEQ_LEN  64
#define GROUP_M  8
#define INV_TAU  10.0f      // 1 / 0.1

extern "C" __global__ void __launch_bounds__(256)
nbest_attn_kernel(const float* __restrict__ log_theta,
                  const int*   __restrict__ ids,
                  float*       __restrict__ out,
                  int N)                     // N = B*M sequences
{
    extern __shared__ float smem[];          // TBL floats (~125 KiB of 320 KB LDS)

    // ---- Stage the unigram table into LDS with the Tensor Data Mover ----
#if __has_builtin(__builtin_amdgcn_tensor_load_to_lds) && \
    __has_builtin(__builtin_amdgcn_s_wait_tensorcnt)
    if (threadIdx.x == 0) {                  // TDM ignores EXEC; one wave issues once
        uint32_t lds_off = __builtin_amdgcn_groupstaticsize(); // dynamic-LDS base
        uint64_t ga = (uint64_t)(uintptr_t)log_theta;

        // D# group 0: count=1 | lds_addr | global_addr[56:0] | type=2
        u32x4 g0;
        g0[0] = 1u;
        g0[1] = lds_off;
        g0[2] = (uint32_t)ga;
        g0[3] = (uint32_t)((ga >> 32) & 0x1FFFFFFull) | (2u << 30);

        // D# group 1: data_size=4B, tensor_dim0=tile_dim0=TBL, dims1=1 (1-row tile)
        i32x8 g1;
        g1[0] = (int)(2u << 16);                         // data_size = 2 (4 bytes)
        g1[1] = (int)((TBL & 0xFFFFu) << 16);            // tensor_dim0[15:0]
        g1[2] = (int)((TBL >> 16) | (1u << 16));         // tensor_dim0[31:16] | tensor_dim1[15:0]=1
        g1[3] = (int)((TBL & 0xFFFFu) << 16);            // tensor_dim1[31:16]=0 | tile_dim0
        g1[4] = 1;                                       // tile_dim1=1, tile_dim2=0 (unused)
        g1[5] = TBL;                                     // tensor_dim0_stride[31:0]
        g1[6] = 0;
        g1[7] = 0;

        i32x4 gz  = {0, 0, 0, 0};                        // groups 2/3 unused (<=2D)
        i32x8 gz8 = {0, 0, 0, 0, 0, 0, 0, 0};            // aux group (clang-23 6-arg form)
        __builtin_amdgcn_tensor_load_to_lds(g0, g1, gz, gz, gz8, 0);
        __builtin_amdgcn_s_wait_tensorcnt(0);            // TENSORcnt==0 before barrier
    }
    __syncthreads();
#else
    for (int i = threadIdx.x; i < TBL; i += blockDim.x) smem[i] = log_theta[i];
    __syncthreads();
#endif

    // ---- Stream ids, gather from LDS, 8-lane group softmaxes ----
    const int stride = gridDim.x * blockDim.x;
    for (int s = blockIdx.x * blockDim.x + threadIdx.x; s < N; s += stride) {
        const int4* __restrict__ row = (const int4*)(ids + (size_t)s * SEQ_LEN);

        // prefetch the next grid-stride row while we gather this one
        if (s + stride < N)
            __builtin_prefetch(ids + (size_t)(s + stride) * SEQ_LEN, 0, 1);

        float acc = 0.0f;                    // logPs for this sequence
#pragma unroll
        for (int j = 0; j < SEQ_LEN / 4; ++j) {   // 16x global_load_b128 per thread
            int4 v = row[j];
            acc += smem[v.x];
            acc += smem[v.y];
            acc += smem[v.z];
            acc += smem[v.w];
        }

        // m=8 hypotheses occupy 8 aligned lanes of the wave32
        float mx = acc;
        mx = fmaxf(mx, __shfl_xor(mx, 1, GROUP_M));
        mx = fmaxf(mx, __shfl_xor(mx, 2, GROUP_M));
        mx = fmaxf(mx, __shfl_xor(mx, 4, GROUP_M));

        float e1 = expf(acc - mx);               // plain softmax numerator
        float s1 = e1;
        s1 += __shfl_xor(s1, 1, GROUP_M);
        s1 += __shfl_xor(s1, 2, GROUP_M);
        s1 += __shfl_xor(s1, 4, GROUP_M);

        float e2 = expf((acc - mx) * INV_TAU);   // temperature softmax numerator
        float s2 = e2;
        s2 += __shfl_xor(s2, 1, GROUP_M);
        s2 += __shfl_xor(s2, 2, GROUP_M);
        s2 += __shfl_xor(s2, 4, GROUP_M);

        out[s]         = e2 / s2;                // attn
        out[N + s]     = e1 / s1;                // softattn
        out[2 * N + s] = acc;                    // logPs
    }
}

extern "C" void kernel_launch(void* const* d_in, const int* in_sizes, int n_in,
                              void* d_out, int out_size, void* d_ws, size_t ws_size,
                              hipStream_t stream) {
    (void)n_in; (void)out_size; (void)d_ws; (void)ws_size;

    const float* log_theta = (const float*)d_in[0];   // [V+2] f32
    const int*   ids       = (const int*)d_in[1];     // [B*M, 64] i32
    float*       out       = (float*)d_out;           // attn | softattn | logPs

    const int N = in_sizes[1] / SEQ_LEN;              // B*M sequences

    const int threads = 256;                          // 8 wave32s / block
    int blocks = (N + threads - 1) / threads;
    if (blocks > 512) blocks = 512;                   // grid-stride; bounds LDS staging traffic
    const size_t shmem = (size_t)TBL * sizeof(float); // ~125 KiB dynamic LDS

    hipLaunchKernelGGL(nbest_attn_kernel, dim3(blocks), dim3(threads), shmem, stream,
                       log_theta, ids, out, N);
}